// MaSIF_ligand_site_69389491634308
// MI455X (gfx1250) — compile-verified
//
#include <hip/hip_runtime.h>

#define K_G   80
#define NFEAT 5
#define NROT  16
#define VPP   200
#define BPATCH 2048
#define EPS_  1e-5f
#define TWO_PI_ 6.283185307179586f

typedef float v2f __attribute__((ext_vector_type(2)));
typedef float v8f __attribute__((ext_vector_type(8)));

// ---------------------------------------------------------------------------
// Kernel 1: per-(patch, feature) gaussian descriptors for all 16 rotations.
// One block per patch. 256 threads = 16 rotations x 16 lanes; each thread
// owns 5 gaussian kernels (stride 16) and reduces over the 200 vertices.
// desc[(b*16 + r)*80 + k] = sum_v g*m*feat / (sum_v g*m + eps)
// ---------------------------------------------------------------------------
__global__ void __launch_bounds__(256) masif_desc_kernel(
    const float* __restrict__ rho,    // [B,V]
    const float* __restrict__ theta,  // [B,V]
    const float* __restrict__ mask,   // [B,V,1]
    const float* __restrict__ feat,   // [B,V,F]
    const float* __restrict__ mu_r,   // [K]  (per-feature slice)
    const float* __restrict__ mu_t,   // [K]
    const float* __restrict__ sg_r,   // [K]
    const float* __restrict__ sg_t,   // [K]
    float* __restrict__ desc,         // [B*NROT, K]
    int f)
{
  __shared__ float s_rho[VPP], s_th[VPP], s_m[VPP], s_f[VPP];
  const int b   = blockIdx.x;
  const int tid = threadIdx.x;

  for (int v = tid; v < VPP; v += 256) {
    const int gi = b * VPP + v;
    s_rho[v] = rho[gi];
    s_th[v]  = theta[gi];
    s_m[v]   = mask[gi];
    s_f[v]   = feat[gi * NFEAT + f];
  }
  __syncthreads();

  const int r = tid >> 4;                            // rotation 0..15
  const int j = tid & 15;                            // k-lane   0..15
  const float roff = (float)r * (TWO_PI_ / (float)NROT);

  float mr[5], mt[5], ar[5], at[5];
  #pragma unroll
  for (int i = 0; i < 5; ++i) {
    const int k = j + 16 * i;
    mr[i] = mu_r[k];
    mt[i] = mu_t[k];
    const float sr = sg_r[k]; ar[i] = 1.0f / (sr * sr + EPS_);
    const float st = sg_t[k]; at[i] = 1.0f / (st * st + EPS_);
  }

  float num[5] = {0.f,0.f,0.f,0.f,0.f};
  float den[5] = {0.f,0.f,0.f,0.f,0.f};

  for (int v = 0; v < VPP; ++v) {
    const float rv = s_rho[v];
    float tv = s_th[v] + roff;                       // theta in [0,2pi) + rot
    tv = (tv >= TWO_PI_) ? tv - TWO_PI_ : tv;        // == mod(.., 2pi)
    const float mv = s_m[v];
    const float wv = mv * s_f[v];
    #pragma unroll
    for (int i = 0; i < 5; ++i) {
      const float dr = rv - mr[i];
      const float dt = tv - mt[i];
      const float q  = fmaf(dt * dt, at[i], dr * dr * ar[i]);
      const float g  = __expf(-q);                   // one v_exp_f32 per gaussian
      den[i] = fmaf(g, mv, den[i]);
      num[i] = fmaf(g, wv, num[i]);
    }
  }

  const int row = b * NROT + r;
  #pragma unroll
  for (int i = 0; i < 5; ++i)
    desc[row * K_G + j + 16 * i] = num[i] / (den[i] + EPS_);
}

// ---------------------------------------------------------------------------
// Kernel 2: conv = desc @ W_conv[f], then max over the 16 rotations (the
// M-dimension of the 16x16 WMMA tile), +bias, relu, scatter into S[b, k*5+f].
// One block per patch; 5 waves, each owning a 16-column tile of K'=80.
// ---------------------------------------------------------------------------
__global__ void __launch_bounds__(160) masif_conv_rotmax_kernel(
    const float* __restrict__ desc,  // [B*16, 80]
    const float* __restrict__ Wc,    // [80,80]   (feature slice)
    const float* __restrict__ bc,    // [80]      (feature slice)
    float* __restrict__ S,           // [B, 400]
    int f)
{
  const int b     = blockIdx.x;
  const int wave  = threadIdx.x >> 5;   // 0..4
  const int lane  = threadIdx.x & 31;
  const int lhalf = lane >> 4;          // 0/1
  const int l16   = lane & 15;
  const int n     = wave * 16 + l16;    // output column

  v8f c = {};
  const float* arow = desc + (size_t)(b * 16 + l16) * K_G;  // row = rotation
  #pragma unroll
  for (int kk = 0; kk < K_G; kk += 4) {
    const int ka = kk + 2 * lhalf;
    v2f a = *(const v2f*)(arow + ka);            // A[M=l16][ka..ka+1]
    v2f bb;
    bb.x = Wc[ka * K_G + n];                     // B[ka  ][n]
    bb.y = Wc[(ka + 1) * K_G + n];               // B[ka+1][n]
    c = __builtin_amdgcn_wmma_f32_16x16x4_f32(false, a, false, bb,
                                              (short)0, c, false, false);
  }

  // max over rotations = max over M: 8 accumulator regs + cross-half shuffle
  float m = c[0];
  #pragma unroll
  for (int i = 1; i < 8; ++i) m = fmaxf(m, c[i]);
  m = fmaxf(m, __shfl_xor(m, 16, 32));
  m = fmaxf(m + bc[n], 0.0f);                    // bias commutes with max
  if (lhalf == 0)
    S[(size_t)b * (K_G * NFEAT) + n * NFEAT + f] = m;  // stack(axis=2) layout
}

// ---------------------------------------------------------------------------
// Kernel 3: FC1  [2048x400] x [400x80] + bias, relu -> T[2048x80]
// 128 blocks (16-row tiles) x 5 waves (16-col tiles), 100 K-steps of 4.
// ---------------------------------------------------------------------------
__global__ void __launch_bounds__(160) masif_fc1_kernel(
    const float* __restrict__ S,    // [2048, 400]
    const float* __restrict__ W1,   // [400, 80]
    const float* __restrict__ b1,   // [80]
    float* __restrict__ T)          // [2048, 80]
{
  const int mt    = blockIdx.x;         // row tile 0..127
  const int wave  = threadIdx.x >> 5;
  const int lane  = threadIdx.x & 31;
  const int lhalf = lane >> 4;
  const int l16   = lane & 15;
  const int n     = wave * 16 + l16;

  v8f c = {};
  const float* arow = S + (size_t)(mt * 16 + l16) * (K_G * NFEAT);
  for (int kk = 0; kk < K_G * NFEAT; kk += 4) {
    const int ka = kk + 2 * lhalf;
    v2f a = *(const v2f*)(arow + ka);
    v2f bb;
    bb.x = W1[ka * K_G + n];
    bb.y = W1[(ka + 1) * K_G + n];
    c = __builtin_amdgcn_wmma_f32_16x16x4_f32(false, a, false, bb,
                                              (short)0, c, false, false);
  }
  const float bias = b1[n];
  #pragma unroll
  for (int i = 0; i < 8; ++i) {
    const int row = mt * 16 + i + 8 * lhalf;     // C layout: VGPR i -> M=i(+8)
    T[(size_t)row * K_G + n] = fmaxf(c[i] + bias, 0.0f);
  }
}

// ---------------------------------------------------------------------------
// Kernel 4: FC2  [2048x80] x [80x5] + bias, relu  (tiny -> plain VALU)
// ---------------------------------------------------------------------------
__global__ void __launch_bounds__(256) masif_fc2_kernel(
    const float* __restrict__ T, const float* __restrict__ W2,
    const float* __restrict__ b2, float* __restrict__ out)
{
  const int idx = blockIdx.x * blockDim.x + threadIdx.x;
  if (idx >= BPATCH * NFEAT) return;
  const int b = idx / NFEAT, n = idx % NFEAT;
  float acc = b2[n];
  const float* trow = T + (size_t)b * K_G;
  #pragma unroll 8
  for (int k = 0; k < K_G; ++k)
    acc = fmaf(trow[k], W2[k * NFEAT + n], acc);
  out[idx] = fmaxf(acc, 0.0f);
}

// ---------------------------------------------------------------------------
extern "C" void kernel_launch(void* const* d_in, const int* in_sizes, int n_in,
                              void* d_out, int out_size, void* d_ws, size_t ws_size,
                              hipStream_t stream) {
  const float* input_feat  = (const float*)d_in[0];
  const float* rho         = (const float*)d_in[1];
  const float* theta       = (const float*)d_in[2];
  const float* mask        = (const float*)d_in[3];
  const float* mu_rho      = (const float*)d_in[4];
  const float* mu_theta    = (const float*)d_in[5];
  const float* sigma_rho   = (const float*)d_in[6];
  const float* sigma_theta = (const float*)d_in[7];
  const float* W_conv      = (const float*)d_in[8];
  const float* b_conv      = (const float*)d_in[9];
  const float* FC1_W       = (const float*)d_in[10];
  const float* FC1_b       = (const float*)d_in[11];
  const float* FC2_W       = (const float*)d_in[12];
  const float* FC2_b       = (const float*)d_in[13];
  float* out = (float*)d_out;

  // workspace layout (floats): desc[32768*80] | S[2048*400] | T[2048*80]
  float* desc = (float*)d_ws;
  float* S    = desc + (size_t)BPATCH * NROT * K_G;
  float* T    = S    + (size_t)BPATCH * K_G * NFEAT;

  for (int f = 0; f < NFEAT; ++f) {
    masif_desc_kernel<<<BPATCH, 256, 0, stream>>>(
        rho, theta, mask, input_feat,
        mu_rho + f * K_G, mu_theta + f * K_G,
        sigma_rho + f * K_G, sigma_theta + f * K_G, desc, f);
    masif_conv_rotmax_kernel<<<BPATCH, 160, 0, stream>>>(
        desc, W_conv + (size_t)f * K_G * K_G, b_conv + f * K_G, S, f);
  }
  masif_fc1_kernel<<<BPATCH / 16, 160, 0, stream>>>(S, FC1_W, FC1_b, T);
  masif_fc2_kernel<<<(BPATCH * NFEAT + 255) / 256, 256, 0, stream>>>(T, FC2_W, FC2_b, out);
}